// GNNEncoder_26422638805245
// MI455X (gfx1250) — compile-verified
//
#include <hip/hip_runtime.h>
#include <hip/hip_bf16.h>
#include <math.h>

// Problem constants (from reference)
#define NN   50000
#define EE   800000
#define DIN  128
#define HID  64
#define NH   4
#define HD   256          // HID*NH
#define ED   5

typedef __attribute__((ext_vector_type(2))) float v2f;
typedef __attribute__((ext_vector_type(8))) float v8f;
typedef __attribute__((ext_vector_type(4))) int   v4i;

typedef __attribute__((address_space(1))) v4i g_v4i;
typedef __attribute__((address_space(3))) v4i l_v4i;

// ---- optional gfx1250 async global->LDS path (ASYNCcnt tracked) ------------
#if defined(__has_builtin)
#if __has_builtin(__builtin_amdgcn_global_load_async_to_lds_b128) && \
    __has_builtin(__builtin_amdgcn_s_wait_asynccnt)
#define USE_ASYNC_LDS 1
#endif
#endif

__device__ inline void ld_tile_b128(const float* gsrc, float* ldst) {
#ifdef USE_ASYNC_LDS
  __builtin_amdgcn_global_load_async_to_lds_b128(
      (g_v4i*)(void*)gsrc, (l_v4i*)(void*)ldst, 0, 0);
#else
  *(float4*)ldst = *(const float4*)gsrc;
#endif
}

// ---------------------------------------------------------------------------
// fp32 WMMA GEMM v3 (double-buffered):  C[M,NC] = A[M,K]@W[K,NC] + bias,
// M == NN.  Block: 256 threads (8 waves). Block tile 128(M) x 64(N);
// K chunked by 32, ping-pong LDS buffers:
//   issue async stage of chunk k+1  ->  s_wait_asynccnt(6) drains chunk k
//   (async loads complete in order; 6 async instrs per stage: 4 A + 2 B)
//   -> barrier -> 32 WMMAs on chunk k -> barrier (frees buffer for k+2).
//  - B chunk (32x64) LDS stride 72, A chunk (128x32) LDS stride 36:
//    bank-conflict free for the f32 16x16x4 fragment reads, b128 aligned.
//  - wave w owns rows [w*16,w*16+16) x 64 cols -> 4 v8f accumulators.
// Requires K%32==0, NC%64==0 (true: K in {128,256}, NC in {64,256}).
// ---------------------------------------------------------------------------
#define BK    32
#define APAD  36
#define BPAD  72

__device__ inline void stage_chunk(const float* __restrict__ A,
                                   const float* __restrict__ W,
                                   int K, int NC, int m0, int n0, int k0,
                                   float* __restrict__ sa,
                                   float* __restrict__ sb, int tid) {
  // A chunk: 128 x 32 floats = 1024 float4, 4 per thread
#pragma unroll
  for (int ii = 0; ii < 4; ++ii) {
    const int i  = tid + ii * 256;
    const int r  = i >> 3;                // 0..127
    const int c4 = (i & 7) * 4;           // 0..28
    const int gr = (m0 + r < NN) ? (m0 + r) : (NN - 1);
    ld_tile_b128(A + (long)gr * K + k0 + c4, &sa[r * APAD + c4]);
  }
  // B chunk: 32 x 64 floats = 512 float4, 2 per thread
#pragma unroll
  for (int ii = 0; ii < 2; ++ii) {
    const int i  = tid + ii * 256;
    const int r  = i >> 4;                // 0..31
    const int c4 = (i & 15) * 4;          // 0..60
    ld_tile_b128(W + (long)(k0 + r) * NC + n0 + c4, &sb[r * BPAD + c4]);
  }
}

__global__ __launch_bounds__(256)
void wmma_gemm_f32(const float* __restrict__ A, const float* __restrict__ W,
                   const float* __restrict__ bias, float* __restrict__ C,
                   int K, int NC) {
  __shared__ float sA[2][128 * APAD];
  __shared__ float sB[2][BK * BPAD];

  const int m0 = blockIdx.x * 128;
  const int n0 = blockIdx.y * 64;
  const int tid  = threadIdx.x;
  const int wv   = tid >> 5;
  const int lane = tid & 31;
  const int hl   = lane >> 4;       // lane-half selects K-pair
  const int lm   = lane & 15;

  v8f acc[4] = {{}, {}, {}, {}};

  // prologue: stage chunk 0 into buffer 0
  stage_chunk(A, W, K, NC, m0, n0, 0, sA[0], sB[0], tid);

  const int nch = K / BK;
  for (int kc = 0; kc < nch; ++kc) {
    const int cur = kc & 1;
    if (kc + 1 < nch) {
      // overlap: issue next chunk's async loads into the other buffer,
      // then wait only for the current chunk's 6 async instructions.
      stage_chunk(A, W, K, NC, m0, n0, (kc + 1) * BK, sA[cur ^ 1], sB[cur ^ 1], tid);
#ifdef USE_ASYNC_LDS
      __builtin_amdgcn_s_wait_asynccnt(6);
#endif
    } else {
#ifdef USE_ASYNC_LDS
      __builtin_amdgcn_s_wait_asynccnt(0);
#endif
    }
    __syncthreads();   // current chunk visible to all waves

    const float* __restrict__ sa = sA[cur];
    const float* __restrict__ sb = sB[cur];
    // 8 k-steps x 4 n-subtiles = 32 WMMAs per chunk
#pragma unroll
    for (int kk = 0; kk < BK; kk += 4) {
      const int ka = kk + 2 * hl;
      v2f a;
      a.x = sa[(wv * 16 + lm) * APAD + ka];
      a.y = sa[(wv * 16 + lm) * APAD + ka + 1];
#pragma unroll
      for (int j = 0; j < 4; ++j) {
        v2f b;
        b.x = sb[ka * BPAD + j * 16 + lm];
        b.y = sb[(ka + 1) * BPAD + j * 16 + lm];
        acc[j] = __builtin_amdgcn_wmma_f32_16x16x4_f32(
            false, a, false, b, (short)0, acc[j], false, false);
      }
    }
    __syncthreads();   // all reads done before buffer is re-staged
  }

#pragma unroll
  for (int j = 0; j < 4; ++j) {
    const float bv = bias[n0 + j * 16 + lm];
#pragma unroll
    for (int v = 0; v < 8; ++v) {
      const int row = m0 + wv * 16 + v + 8 * hl;
      if (row < NN)
        C[(long)row * NC + n0 + j * 16 + lm] = acc[j][v] + bv;
    }
  }
}

// ---------------------------------------------------------------------------
// float atomic max via signed/unsigned trick (valid for -inf init)
// ---------------------------------------------------------------------------
__device__ inline void atomicMaxF(float* addr, float val) {
  if (val >= 0.0f) atomicMax((int*)addr, __float_as_int(val));
  else             atomicMin((unsigned int*)addr, __float_as_uint(val));
}

// ---------------------------------------------------------------------------
// init: zero agg[N*HD], den[N*NH]; set nmax[N*NH] = -inf
// ---------------------------------------------------------------------------
__global__ void init_bufs(float* __restrict__ agg, float* __restrict__ nmax,
                          float* __restrict__ den) {
  const long i = (long)blockIdx.x * blockDim.x + threadIdx.x;
  if (i < (long)NN * HD) agg[i] = 0.0f;
  if (i < (long)NN * NH) { nmax[i] = -__builtin_inff(); den[i] = 0.0f; }
}

// ---------------------------------------------------------------------------
// Pass A: per-edge attention logits + segment max.  One wave per edge.
// lane -> 8 contiguous channels hc=[8*lane, 8*lane+8) (all in head lane/8).
// ---------------------------------------------------------------------------
__global__ __launch_bounds__(256)
void edge_logit(const int* __restrict__ src, const int* __restrict__ dst,
                const float* __restrict__ ea, const float* __restrict__ We,
                const float* __restrict__ att, const float* __restrict__ tempP,
                const float* __restrict__ XL, const float* __restrict__ XR,
                float* __restrict__ logw, float* __restrict__ nmax) {
  const int wave = threadIdx.x >> 5;
  const int lane = threadIdx.x & 31;
  const long e = (long)blockIdx.x * 8 + wave;
  if (e >= EE) return;
  const int s = src[e], d = dst[e];

  const float a0 = ea[e * ED + 0];
  const float a1 = ea[e * ED + 1];
  const float a2 = ea[e * ED + 2];
  const float a3 = ea[e * ED + 3];
  const float a4 = tempP[0] / fmaxf(a3, 1e-6f);   // physics: temp*PHYS_W/|Z|

  const float* __restrict__ xls = XL + (long)s * HD;
  const float* __restrict__ xrd = XR + (long)d * HD;

  float acc = 0.0f;
  const int base = lane * 8;
#pragma unroll
  for (int j = 0; j < 8; ++j) {
    const int hc = base + j;
    float ev = a0 * We[hc] + a1 * We[HD + hc] + a2 * We[2 * HD + hc] +
               a3 * We[3 * HD + hc] + a4 * We[4 * HD + hc];
    float m = xls[hc] + xrd[hc] + ev;
    m = (m > 0.0f) ? m : 0.2f * m;             // leaky_relu(0.2)
    acc += m * att[hc];                        // att flat [h*HID + c] == [hc]
  }
  // reduce over 8 lanes of the same head
  acc += __shfl_xor(acc, 1, 32);
  acc += __shfl_xor(acc, 2, 32);
  acc += __shfl_xor(acc, 4, 32);
  if ((lane & 7) == 0) {
    const int h = lane >> 3;
    logw[e * NH + h] = acc;
    atomicMaxF(&nmax[(long)d * NH + h], acc);
  }
}

// ---------------------------------------------------------------------------
// Pass B: w = exp(logit - max[dst]); den[dst] += w.  One thread per (edge,h).
// ---------------------------------------------------------------------------
__global__ void edge_exp(const int* __restrict__ dst, const float* __restrict__ nmax,
                         float* __restrict__ logw, float* __restrict__ den) {
  const long i = (long)blockIdx.x * blockDim.x + threadIdx.x;
  if (i >= (long)EE * NH) return;
  const long e = i >> 2;
  const int h = (int)(i & 3);
  const int d = dst[e];
  const float w = expf(logw[i] - nmax[(long)d * NH + h]);
  logw[i] = w;
  atomicAdd(&den[(long)d * NH + h], w);
}

// ---------------------------------------------------------------------------
// Pass C: agg[dst] += XL[src] * (w / (den[dst]+1e-16)).  One wave per edge.
// ---------------------------------------------------------------------------
__global__ __launch_bounds__(256)
void edge_scatter(const int* __restrict__ src, const int* __restrict__ dst,
                  const float* __restrict__ XL, const float* __restrict__ logw,
                  const float* __restrict__ den, float* __restrict__ agg) {
  const int wave = threadIdx.x >> 5;
  const int lane = threadIdx.x & 31;
  const long e = (long)blockIdx.x * 8 + wave;
  if (e >= EE) return;
  const int s = src[e], d = dst[e];
  const int h = lane >> 3;
  const float a = logw[e * NH + h] / (den[(long)d * NH + h] + 1e-16f);
  const float* __restrict__ xls = XL + (long)s * HD;
  float* __restrict__ ad = agg + (long)d * HD;
  const int base = lane * 8;
#pragma unroll
  for (int j = 0; j < 8; ++j)
    atomicAdd(&ad[base + j], xls[base + j] * a);
}

// ---------------------------------------------------------------------------
// Post (concat layers 0/1): h = maybe_elu(LN(agg + bias) + res). Wave/node.
// Safe when out == res (per-thread read-before-write, disjoint elements).
// ---------------------------------------------------------------------------
__global__ __launch_bounds__(256)
void post_concat(const float* __restrict__ agg, const float* __restrict__ bias,
                 const float* __restrict__ lnw, const float* __restrict__ lnb,
                 const float* __restrict__ res, float* __restrict__ out, int doElu) {
  const int wave = threadIdx.x >> 5;
  const int lane = threadIdx.x & 31;
  const long n = (long)blockIdx.x * 8 + wave;
  if (n >= NN) return;
  const float* __restrict__ g = agg + n * HD;

  float v[8];
  float s = 0.0f;
#pragma unroll
  for (int j = 0; j < 8; ++j) {
    const int c = lane * 8 + j;
    v[j] = g[c] + bias[c];
    s += v[j];
  }
#pragma unroll
  for (int off = 16; off >= 1; off >>= 1) s += __shfl_xor(s, off, 32);
  const float mu = s * (1.0f / HD);

  float q = 0.0f;
#pragma unroll
  for (int j = 0; j < 8; ++j) { const float t = v[j] - mu; q += t * t; }
#pragma unroll
  for (int off = 16; off >= 1; off >>= 1) q += __shfl_xor(q, off, 32);
  const float inv = rsqrtf(q * (1.0f / HD) + 1e-5f);

#pragma unroll
  for (int j = 0; j < 8; ++j) {
    const int c = lane * 8 + j;
    float y = (v[j] - mu) * inv * lnw[c] + lnb[c] + res[n * HD + c];
    if (doElu) y = (y > 0.0f) ? y : (expf(y) - 1.0f);
    out[n * HD + c] = y;
  }
}

// ---------------------------------------------------------------------------
// Post (layer 2, head-mean): out = LN(mean_h(agg) + bias) + res. Wave/node.
// ---------------------------------------------------------------------------
__global__ __launch_bounds__(256)
void post_mean(const float* __restrict__ agg, const float* __restrict__ bias,
               const float* __restrict__ lnw, const float* __restrict__ lnb,
               const float* __restrict__ res, float* __restrict__ out) {
  const int wave = threadIdx.x >> 5;
  const int lane = threadIdx.x & 31;
  const long n = (long)blockIdx.x * 8 + wave;
  if (n >= NN) return;
  const float* __restrict__ g = agg + n * HD;

  float v[2];
  float s = 0.0f;
#pragma unroll
  for (int j = 0; j < 2; ++j) {
    const int c = lane * 2 + j;
    v[j] = 0.25f * (g[c] + g[HID + c] + g[2 * HID + c] + g[3 * HID + c]) + bias[c];
    s += v[j];
  }
#pragma unroll
  for (int off = 16; off >= 1; off >>= 1) s += __shfl_xor(s, off, 32);
  const float mu = s * (1.0f / HID);

  float q = 0.0f;
#pragma unroll
  for (int j = 0; j < 2; ++j) { const float t = v[j] - mu; q += t * t; }
#pragma unroll
  for (int off = 16; off >= 1; off >>= 1) q += __shfl_xor(q, off, 32);
  const float inv = rsqrtf(q * (1.0f / HID) + 1e-5f);

#pragma unroll
  for (int j = 0; j < 2; ++j) {
    const int c = lane * 2 + j;
    out[n * HID + c] = (v[j] - mu) * inv * lnw[c] + lnb[c] + res[n * HID + c];
  }
}

// ---------------------------------------------------------------------------
// Host-side launch
// ---------------------------------------------------------------------------
extern "C" void kernel_launch(void* const* d_in, const int* in_sizes, int n_in,
                              void* d_out, int out_size, void* d_ws, size_t ws_size,
                              hipStream_t stream) {
  (void)in_sizes; (void)n_in; (void)out_size; (void)ws_size;

  const float* x  = (const float*)d_in[0];
  const int*   ei = (const int*)d_in[1];
  const float* ea = (const float*)d_in[2];
  const int* src = ei;
  const int* dst = ei + EE;

  // per-layer params: base = 3 + i*10 -> Wl, bl, Wr, br, We, att, temp, bias, lnw, lnb
  const float* P[3][10];
  for (int i = 0; i < 3; ++i)
    for (int j = 0; j < 10; ++j)
      P[i][j] = (const float*)d_in[3 + i * 10 + j];
  const float* pw0 = (const float*)d_in[33];
  const float* pb0 = (const float*)d_in[34];
  const float* pw2 = (const float*)d_in[35];
  const float* pb2 = (const float*)d_in[36];

  // workspace carve-up (floats)
  float* w      = (float*)d_ws;
  float* f_xl   = w;  w += (size_t)NN * HD;
  float* f_xr   = w;  w += (size_t)NN * HD;
  float* f_agg  = w;  w += (size_t)NN * HD;
  float* f_h    = w;  w += (size_t)NN * HD;
  float* f_res  = w;  w += (size_t)NN * HD;
  float* f_logw = w;  w += (size_t)EE * NH;
  float* f_max  = w;  w += (size_t)NN * NH;
  float* f_den  = w;  w += (size_t)NN * NH;

  const dim3 gemmBlk(256);
  const int  mTiles = (NN + 127) / 128;
  const dim3 edgeGrid((EE + 7) / 8), blk256(256);
  const dim3 nodeGrid((NN + 7) / 8);
  const int initGrid = (NN * HD + 255) / 256;
  const int expGrid  = ((long)EE * NH + 255) / 256;

  const float* feat = x;
  int K = DIN;

  for (int L = 0; L < 3; ++L) {
    // node GEMMs: XL = feat@Wl + bl ; XR = feat@Wr + br
    dim3 g(mTiles, HD / 64);
    wmma_gemm_f32<<<g, gemmBlk, 0, stream>>>(feat, P[L][0], P[L][1], f_xl, K, HD);
    wmma_gemm_f32<<<g, gemmBlk, 0, stream>>>(feat, P[L][2], P[L][3], f_xr, K, HD);
    // residual projections
    if (L == 0)
      wmma_gemm_f32<<<dim3(mTiles, HD / 64), gemmBlk, 0, stream>>>(x, pw0, pb0, f_res, DIN, HD);
    if (L == 2)
      wmma_gemm_f32<<<dim3(mTiles, HID / 64), gemmBlk, 0, stream>>>(f_h, pw2, pb2, f_res, HD, HID);

    // segment softmax + aggregation
    init_bufs<<<initGrid, blk256, 0, stream>>>(f_agg, f_max, f_den);
    edge_logit<<<edgeGrid, blk256, 0, stream>>>(src, dst, ea, P[L][4], P[L][5],
                                                P[L][6], f_xl, f_xr, f_logw, f_max);
    edge_exp<<<expGrid, blk256, 0, stream>>>(dst, f_max, f_logw, f_den);
    edge_scatter<<<edgeGrid, blk256, 0, stream>>>(src, dst, f_xl, f_logw, f_den, f_agg);

    // post: LN + residual (+ elu)
    if (L == 0) {
      post_concat<<<nodeGrid, blk256, 0, stream>>>(f_agg, P[0][7], P[0][8], P[0][9],
                                                   f_res, f_h, /*elu=*/1);
      feat = f_h; K = HD;
    } else if (L == 1) {
      post_concat<<<nodeGrid, blk256, 0, stream>>>(f_agg, P[1][7], P[1][8], P[1][9],
                                                   f_h, f_h, /*elu=*/1);   // identity residual, in-place
      feat = f_h; K = HD;
    } else {
      post_mean<<<nodeGrid, blk256, 0, stream>>>(f_agg, P[2][7], P[2][8], P[2][9],
                                                 f_res, (float*)d_out);
    }
  }
}